// IntegralSelfAttention_75196287418665
// MI455X (gfx1250) — compile-verified
//
#include <hip/hip_runtime.h>

// ---------------------------------------------------------------------------
// IntegralSelfAttention on MI455X (gfx1250, wave32).
// fp32 end-to-end using V_WMMA_F32_16X16X4_F32. Flash-style attention so the
// (B,H,N,N)=268MB logits tensor never touches HBM (total traffic ~80MB).
// ---------------------------------------------------------------------------

#define D_MODEL   512
#define NHEADS    8
#define HEAD_DIM  64
#define SEQ       2048
#define BATCH     2
#define ROWS      (BATCH * SEQ)          // 4096
#define LN_EPS    1e-5f
// 1 / (NUM_SIGNALS * sqrt(D_H)) = 1 / (8 * sqrt(8))
#define ATT_SCALE 0.04419417382415922f

typedef __attribute__((ext_vector_type(2))) float v2f;
typedef __attribute__((ext_vector_type(8))) float v8f;

// D = A(16x4) * B(4x16) + C(16x16), fp32 WMMA
__device__ __forceinline__ v8f wmma4(v2f a, v2f b, v8f c) {
  return __builtin_amdgcn_wmma_f32_16x16x4_f32(
      /*neg_a=*/false, a, /*neg_b=*/false, b,
      /*c_mod=*/(short)0, c, /*reuse_a=*/false, /*reuse_b=*/false);
}

// ---------------------------------------------------------------------------
// LayerNorm: one wave per row of (4096, 512)
// ---------------------------------------------------------------------------
__global__ void isa_ln_kernel(const float* __restrict__ x,
                              const float* __restrict__ g,
                              const float* __restrict__ bb,
                              float* __restrict__ xn) {
  int row  = blockIdx.x * 8 + (threadIdx.x >> 5);
  int lane = threadIdx.x & 31;
  const float* xr = x + (size_t)row * D_MODEL;
  float v[16];
  float s = 0.f;
#pragma unroll
  for (int i = 0; i < 16; ++i) { v[i] = xr[lane + i * 32]; s += v[i]; }
#pragma unroll
  for (int off = 16; off >= 1; off >>= 1) s += __shfl_xor(s, off, 32);
  float mu = s * (1.0f / 512.0f);
  float q = 0.f;
#pragma unroll
  for (int i = 0; i < 16; ++i) { float d = v[i] - mu; q += d * d; }
#pragma unroll
  for (int off = 16; off >= 1; off >>= 1) q += __shfl_xor(q, off, 32);
  float rinv = rsqrtf(q * (1.0f / 512.0f) + LN_EPS);
  float* xo = xn + (size_t)row * D_MODEL;
#pragma unroll
  for (int i = 0; i < 16; ++i) {
    int c = lane + i * 32;
    xo[c] = (v[i] - mu) * rinv * g[c] + bb[c];
  }
}

// ---------------------------------------------------------------------------
// GEMM  Y(MxN) = A(MxK) * W(KxN) + bias  (all row-major, fp32 WMMA)
// One wave computes a 16(M) x 64(N) tile: A-frag reused across 4 accumulators.
// ---------------------------------------------------------------------------
__global__ void isa_gemm_kernel(const float* __restrict__ A,
                                const float* __restrict__ W,
                                const float* __restrict__ bias,
                                float* __restrict__ Y,
                                int M, int N, int K) {
  int w    = blockIdx.x * 8 + (threadIdx.x >> 5);
  int lane = threadIdx.x & 31;
  int ntiles = N >> 6;
  int m0 = (w / ntiles) * 16;
  int n0 = (w % ntiles) * 64;
  if (m0 >= M) return;
  int l16 = lane & 15, half = lane >> 4;

  v8f acc[4] = {};
  const float* arow = A + (size_t)(m0 + l16) * K + (half << 1);
  for (int k0 = 0; k0 < K; k0 += 4) {
    v2f a = *(const v2f*)(arow + k0);   // A[m0+l16][k0 + 2*half .. +1]
#pragma unroll
    for (int j = 0; j < 4; ++j) {
      int col = n0 + j * 16 + l16;
      int kb  = k0 + (half << 1);
      v2f b;
      b.x = W[(size_t)kb * N + col];          // row kb   (v0: k0 / k0+2)
      b.y = W[(size_t)(kb + 1) * N + col];    // row kb+1 (v1: k0+1 / k0+3)
      acc[j] = wmma4(a, b, acc[j]);
    }
  }
#pragma unroll
  for (int j = 0; j < 4; ++j) {
    int col  = n0 + j * 16 + l16;
    float bv = bias[col];
#pragma unroll
    for (int r = 0; r < 8; ++r)
      Y[(size_t)(m0 + r + half * 8) * N + col] = acc[j][r] + bv;
  }
}

// ---------------------------------------------------------------------------
// Flash attention: one wave owns 16 queries x one head (hd=64).
// Online softmax; P staged via LDS to re-layout C/D -> A fragments.
// Output written directly in (B, N, H*hd) layout.
// ---------------------------------------------------------------------------
__global__ void isa_attn_kernel(const float* __restrict__ Q,
                                const float* __restrict__ Km,
                                const float* __restrict__ V,
                                const int* __restrict__ mask,
                                float* __restrict__ O) {
  __shared__ float plds[8][16 * 16];
  int warp = threadIdx.x >> 5;
  int wv   = blockIdx.x * 8 + warp;
  int lane = threadIdx.x & 31;
  int qt = wv & 127;          // 128 query tiles of 16
  int h  = (wv >> 7) & 7;
  int b  = wv >> 10;
  int l16 = lane & 15, half = lane >> 4;
  int qn0 = qt * 16;
  const size_t ld = D_MODEL;

  // Preload 16x64 Q tile as 16 A-fragments (k-steps of 4)
  const float* qbase = Q + ((size_t)(b * SEQ) + qn0) * ld + h * HEAD_DIM;
  v2f qa[16];
#pragma unroll
  for (int ks = 0; ks < 16; ++ks)
    qa[ks] = *(const v2f*)(qbase + (size_t)l16 * ld + ks * 4 + (half << 1));

  v8f o[4] = {};
  float rm[8], rl[8];
#pragma unroll
  for (int r = 0; r < 8; ++r) { rm[r] = -3.0e38f; rl[r] = 0.f; }

  const int*   mrow  = mask + b * SEQ;
  const float* kbase = Km + (size_t)(b * SEQ) * ld + h * HEAD_DIM;
  const float* vbase = V  + (size_t)(b * SEQ) * ld + h * HEAD_DIM;
  float* pl = plds[warp];

  for (int kn0 = 0; kn0 < SEQ; kn0 += 16) {
    // ---- S = Q * K^T  (16x16 tile, K-dim = 64 -> 16 WMMAs) ----
    v8f s = {};
#pragma unroll
    for (int ks = 0; ks < 16; ++ks) {
      // B[kdim][n] = K[key kn0+n][kdim]; rows ks*4+2*half, +1 are contiguous
      v2f bfr = *(const v2f*)(kbase + (size_t)(kn0 + l16) * ld +
                              ks * 4 + (half << 1));
      s = wmma4(qa[ks], bfr, s);
    }

    // ---- scale + key-padding mask (column key = kn0 + l16) ----
    int mk = mrow[kn0 + l16];
#pragma unroll
    for (int r = 0; r < 8; ++r)
      s[r] = (mk != 0) ? s[r] * ATT_SCALE : -10000.0f;

    // ---- online softmax (row = 16 lanes within a half-wave) ----
    float pr[8];
#pragma unroll
    for (int r = 0; r < 8; ++r) {
      float m = s[r];
      m = fmaxf(m, __shfl_xor(m, 1, 16));
      m = fmaxf(m, __shfl_xor(m, 2, 16));
      m = fmaxf(m, __shfl_xor(m, 4, 16));
      m = fmaxf(m, __shfl_xor(m, 8, 16));
      float nm    = fmaxf(rm[r], m);
      float alpha = __expf(rm[r] - nm);
      float p     = __expf(s[r] - nm);
      rm[r] = nm;
      float ps = p;
      ps += __shfl_xor(ps, 1, 16);
      ps += __shfl_xor(ps, 2, 16);
      ps += __shfl_xor(ps, 4, 16);
      ps += __shfl_xor(ps, 8, 16);
      rl[r] = rl[r] * alpha + ps;
      pr[r] = p;
#pragma unroll
      for (int j = 0; j < 4; ++j) o[j][r] *= alpha;
    }

    // ---- P: C/D layout -> LDS -> A fragments (wave-local fence) ----
#pragma unroll
    for (int r = 0; r < 8; ++r)
      pl[(r + half * 8) * 16 + l16] = pr[r];
    __builtin_amdgcn_wave_barrier();
    asm volatile("s_wait_dscnt 0" ::: "memory");
    __builtin_amdgcn_wave_barrier();
    v2f pa[4];
#pragma unroll
    for (int ks = 0; ks < 4; ++ks)
      pa[ks] = *(const v2f*)(pl + l16 * 16 + ks * 4 + (half << 1));

    // ---- O += P(16x16) * V(16x64)  (16 WMMAs) ----
#pragma unroll
    for (int j = 0; j < 4; ++j) {
#pragma unroll
      for (int ks = 0; ks < 4; ++ks) {
        int kb = ks * 4 + (half << 1);   // key row within tile
        v2f bfr;
        bfr.x = vbase[(size_t)(kn0 + kb) * ld + j * 16 + l16];
        bfr.y = vbase[(size_t)(kn0 + kb + 1) * ld + j * 16 + l16];
        o[j] = wmma4(pa[ks], bfr, o[j]);
      }
    }
  }

  // ---- finalize: divide by row sum, write (B,N,C) ----
  float* obase = O + ((size_t)(b * SEQ) + qn0) * ld + h * HEAD_DIM;
#pragma unroll
  for (int r = 0; r < 8; ++r) {
    float inv = (rl[r] > 0.f) ? 1.0f / rl[r] : 0.f;
#pragma unroll
    for (int j = 0; j < 4; ++j)
      obase[(size_t)(r + half * 8) * ld + j * 16 + l16] = o[j][r] * inv;
  }
}

// ---------------------------------------------------------------------------
extern "C" void kernel_launch(void* const* d_in, const int* in_sizes, int n_in,
                              void* d_out, int out_size, void* d_ws, size_t ws_size,
                              hipStream_t stream) {
  const float* x    = (const float*)d_in[0];
  const int*   mask = (const int*)  d_in[1];
  const float* ln_g = (const float*)d_in[2];
  const float* ln_b = (const float*)d_in[3];
  const float* wq   = (const float*)d_in[4];
  const float* bq   = (const float*)d_in[5];
  const float* wk   = (const float*)d_in[6];
  const float* bk   = (const float*)d_in[7];
  const float* wv   = (const float*)d_in[8];
  const float* bv   = (const float*)d_in[9];
  const float* wo   = (const float*)d_in[10];
  const float* bo   = (const float*)d_in[11];
  float* out = (float*)d_out;

  char* ws = (char*)d_ws;
  const size_t sz = (size_t)ROWS * D_MODEL * sizeof(float);  // 8 MB each
  float* xn  = (float*)(ws);
  float* q   = (float*)(ws + 1 * sz);
  float* k   = (float*)(ws + 2 * sz);
  float* v   = (float*)(ws + 3 * sz);
  float* att = (float*)(ws + 4 * sz);

  isa_ln_kernel<<<ROWS / 8, 256, 0, stream>>>(x, ln_g, ln_b, xn);

  // (4096/16) M-tiles * (512/64) N-tiles = 2048 waves = 256 blocks of 8 waves
  const int gblocks = (ROWS / 16) * (D_MODEL / 64) / 8;
  isa_gemm_kernel<<<gblocks, 256, 0, stream>>>(xn, wq, bq, q, ROWS, D_MODEL, D_MODEL);
  isa_gemm_kernel<<<gblocks, 256, 0, stream>>>(xn, wk, bk, k, ROWS, D_MODEL, D_MODEL);
  isa_gemm_kernel<<<gblocks, 256, 0, stream>>>(xn, wv, bv, v, ROWS, D_MODEL, D_MODEL);

  // B*H*(N/16) = 2048 waves = 256 blocks
  isa_attn_kernel<<<(BATCH * NHEADS * (SEQ / 16)) / 8, 256, 0, stream>>>(
      q, k, v, mask, att);

  isa_gemm_kernel<<<gblocks, 256, 0, stream>>>(att, wo, bo, out, ROWS, D_MODEL, D_MODEL);
}